// JAStatic3_79199196938492
// MI455X (gfx1250) — compile-verified
//
#include <hip/hip_runtime.h>
#include <hip/hip_bf16.h>
#include <math.h>

// ---- Reference constants (float32) ----
#define TAU_F      6.25e-8f               // 1/16e6
#define INV_TAU_F  1.6e7f
#define MU0_F      1.2566370614359172e-6f // 4e-7 * pi
#define INV_MU0_F  795774.7154594767f
#define CLIP_F     1.0e8f
#define INV_CLIP_F 1.0e-8f
#define EPS_F      1.0e-7f

typedef float v2f __attribute__((ext_vector_type(2)));

// gfx1250 V_TANH_F32 (TRANS op) via builtin; libm fallback for safety.
__device__ __forceinline__ float fast_tanh(float x) {
#if __has_builtin(__builtin_amdgcn_tanhf)
    return __builtin_amdgcn_tanhf(x);
#elif __has_builtin(__builtin_amdgcn_tanh_f32)
    return __builtin_amdgcn_tanh_f32(x);
#else
    return tanhf(x);
#endif
}

__device__ __forceinline__ float fast_rcp(float x) { return __builtin_amdgcn_rcpf(x); }
__device__ __forceinline__ float fast_rsq(float x) { return __builtin_amdgcn_rsqf(x); }

__device__ __forceinline__ float sgnf(float x) {
    return (x > 0.0f) ? 1.0f : ((x < 0.0f) ? -1.0f : 0.0f);
}

__device__ __forceinline__ float sigmoidf_(float x) {
    return 1.0f / (1.0f + __expf(-x));
}

struct JAParams {
    float a, inv_a, alpha, k, ck, Ms, Ms_over_a, C_edd, C_exc;
};

// One Jiles-Atherton scan step. Critical chain: ~12 cheap VALU ops plus
// v_rsq, 2x v_tanh, 3x v_rcp. The reference's two divides
// (num/den, then /(1+dM_dH)) are fused:  dH_pre = (dB/mu0)*den/(den+num).
__device__ __forceinline__ float ja_step(float H, float Bp, float Bn,
                                         const JAParams& P) {
    const float dB    = (Bn - Bp) * INV_TAU_F;               // dB/dt
    const float M     = fmaf(Bp, INV_MU0_F, -H);             // B/mu0 - H
    const float H_exc = P.C_exc * dB * fast_rsq(fabsf(dB));  // dB/sqrt(|dB|)
    const float H_e   = fmaf(P.alpha, M, H) - P.C_edd * dB - H_exc;

    // coth_stable(H_e / a)
    float x = H_e * P.inv_a;
    if (fabsf(x) < EPS_F) x = EPS_F * sgnf(x);
    const float cth  = fast_rcp(fast_tanh(x));
    const float arHe = P.a * fast_rcp(H_e);                  // a / H_e

    const float M_an    = P.Ms * (cth - arHe);
    const float dMdiff  = M_an - M;
    const float delta   = sgnf(dB);
    const float delta_m = 0.5f * (1.0f + sgnf(dMdiff * dB));
    const float dM_an   = P.Ms_over_a * (1.0f - cth * cth + arHe * arHe);

    const float num = fmaf(delta_m, dMdiff, P.ck * delta * dM_an);
    const float den = fmaf(-P.alpha, num, P.k * delta);

    const float pre   = dB * INV_MU0_F * den * fast_rcp(den + num);
    const float dH_dt = CLIP_F * fast_tanh(pre * INV_CLIP_F);  // softclip
    return fmaf(TAU_F, dH_dt, H);
}

// Each thread owns two ADJACENT columns (2i, 2i+1): two independent
// recurrence chains (deterministic ~2x issue-slot utilization per wave on
// this latency-bound serial scan) moved with a single b64 load / b64 store
// per step. A wave covers 256 contiguous bytes per access -> perfectly
// coalesced. L2 prefetch runs 16 rows (~1600+ cycles) ahead to cover HBM
// latency; the 1-iteration register pipeline covers the resulting L2-hit
// latency. Output is streamed with non-temporal stores so the 192MB L2
// stays available for the B_seq read stream.
__global__ __launch_bounds__(128)
void ja_scan_kernel(const float* __restrict__ H0,
                    const float* __restrict__ Bseq,
                    const float* __restrict__ pMs,
                    const float* __restrict__ pa,
                    const float* __restrict__ palpha,
                    const float* __restrict__ pk,
                    const float* __restrict__ pc,
                    const float* __restrict__ pCedd,
                    const float* __restrict__ pCexc,
                    float* __restrict__ Hout,
                    int T, int Bcols)
{
    const int nPairs = Bcols >> 1;
    const int tid = blockIdx.x * blockDim.x + threadIdx.x;
    if (tid >= nPairs || T < 2) return;

    // ---- parameter transforms (once per thread, off the hot loop) ----
    JAParams P;
    P.a      = 2000.0f * sigmoidf_(pa[0]);
    P.alpha  = 0.01f   * sigmoidf_(palpha[0]);
    P.k      = 1000.0f * sigmoidf_(pk[0]);
    P.Ms     = 6.0e6f  * sigmoidf_(pMs[0]);
    P.C_edd  =           sigmoidf_(pCedd[0]);
    P.C_exc  =           sigmoidf_(pCexc[0]);
    const float c = sigmoidf_(pc[0]);
    P.inv_a     = 1.0f / P.a;
    P.Ms_over_a = P.Ms * P.inv_a;
    P.ck        = c * P.k;

    const size_t rs2 = (size_t)nPairs;          // row stride in float2 units

    v2f Hv = ((const v2f*)H0)[tid];

    const v2f* __restrict__ bp = (const v2f*)Bseq + tid;  // points at loaded row
    v2f Bp = bp[0];
    bp += rs2;
    v2f Bn = bp[0];                              // row 1, one step ahead

    v2f* __restrict__ op = (v2f*)Hout + tid;

    // Main loop: steps t = 1 .. T-2, pipelined load of row t+1.
    for (int t = 1; t < T - 1; ++t) {
        bp += rs2;
        const v2f Bf = bp[0];                    // row t+1 (needed next iter)
        __builtin_prefetch((const float*)(bp + 16 * rs2), 0, 0); // L2, ~16 rows ahead

        Hv.x = ja_step(Hv.x, Bp.x, Bn.x, P);     // two independent chains,
        Hv.y = ja_step(Hv.y, Bp.y, Bn.y, P);     // interleaved by the compiler

        __builtin_nontemporal_store(Hv, op);     // streaming write (th:NT)
        op += rs2;
        Bp = Bn;  Bn = Bf;
    }

    // Final step t = T-1 (no further load needed).
    Hv.x = ja_step(Hv.x, Bp.x, Bn.x, P);
    Hv.y = ja_step(Hv.y, Bp.y, Bn.y, P);
    __builtin_nontemporal_store(Hv, op);
}

extern "C" void kernel_launch(void* const* d_in, const int* in_sizes, int n_in,
                              void* d_out, int out_size, void* d_ws, size_t ws_size,
                              hipStream_t stream) {
    // setup_inputs() dict order:
    // 0: H0 (B,)   1: B_seq (T,B)   2: Ms   3: a   4: alpha
    // 5: k         6: c             7: C_edd 8: C_exc
    const float* H0    = (const float*)d_in[0];
    const float* Bseq  = (const float*)d_in[1];
    const float* pMs   = (const float*)d_in[2];
    const float* pa    = (const float*)d_in[3];
    const float* palph = (const float*)d_in[4];
    const float* pk    = (const float*)d_in[5];
    const float* pc    = (const float*)d_in[6];
    const float* pCedd = (const float*)d_in[7];
    const float* pCexc = (const float*)d_in[8];
    float* Hout = (float*)d_out;

    const int Bcols = in_sizes[0];              // 8192
    const int T     = in_sizes[1] / Bcols;      // 4096

    const int threads = 128;                    // 4 wave32 per workgroup
    const int nPairs  = Bcols / 2;              // 2 adjacent columns per thread
    const int blocks  = (nPairs + threads - 1) / threads;  // 32 WGs

    ja_scan_kernel<<<blocks, threads, 0, stream>>>(
        H0, Bseq, pMs, pa, palph, pk, pc, pCedd, pCexc, Hout, T, Bcols);
}